// HuberEMA_4870492914166
// MI455X (gfx1250) — compile-verified
//
#include <hip/hip_runtime.h>
#include <cstdint>

// Problem dims (fixed by reference): B=32, T=4096, C=512, delta=1.0
#define B_TOTAL 32
#define T_TOTAL 4096
#define C_TOTAL 512

// Tiling: each block = one batch b, 128 channels; 64 threads (2 waves), 2 ch/lane.
#define CC 128          // channels per block
#define TT 32           // timesteps per LDS tile
#define THREADS 64
#define NBUF 3          // triple-buffered async pipeline (48 outstanding <= 63 ASYNCcnt)
#define CHUNKS_PER_THREAD ((TT * CC / 4) / THREADS)   // 16 b128 chunks per thread per tile

typedef float v2f __attribute__((ext_vector_type(2)));

__global__ __launch_bounds__(THREADS) void huber_ema_scan_kernel(
    const float* __restrict__ x,
    const float* __restrict__ logit_alpha,
    float* __restrict__ out)
{
    __shared__ __align__(16) float tile[NBUF][TT * CC];   // 48 KB

    const int lane = threadIdx.x;                          // 0..63
    const int b    = blockIdx.x / (C_TOTAL / CC);
    const int c0   = (blockIdx.x % (C_TOTAL / CC)) * CC;
    const int c    = c0 + 2 * lane;                        // this lane's first channel

    // Per-channel one_minus_alpha (clipped sigmoid), computed once.
    float oma0, oma1;
    {
        float l0 = logit_alpha[c];
        float l1 = logit_alpha[c + 1];
        float a0 = 1.0f / (1.0f + __expf(-l0));
        float a1 = 1.0f / (1.0f + __expf(-l1));
        a0 = fminf(fmaxf(a0, 1.0e-4f), 1.0f - 1.0e-4f);
        a1 = fminf(fmaxf(a1, 1.0e-4f), 1.0f - 1.0e-4f);
        oma0 = 1.0f - a0;
        oma1 = 1.0f - a1;
    }

    // Seed the recurrence with y = x[b,0,c]. The t=0 step then computes
    // r = 0 -> g = med3(0,-1,1) = 0 -> y unchanged, which matches the
    // reference's y_0 = x_0 exactly (no special-case branch needed).
    const size_t seq_base = ((size_t)b * T_TOTAL) * C_TOTAL + (size_t)c;
    float y0, y1;
    {
        const v2f xv0 = *(const v2f*)(x + seq_base);
        y0 = xv0.x;
        y1 = xv0.y;
    }

    // Wave-relative LDS byte offset of the tile buffer (low 32 bits of the
    // generic pointer are the LDS offset on gfx1250).
    const uint32_t lds_base = (uint32_t)(uintptr_t)(&tile[0][0]);
    const uint64_t xbase    = (uint64_t)(uintptr_t)x;

    // Per-lane DMA geometry: flat chunk j = lane + 64*k maps to
    //   tt = (lane>>5) + 2k,  c4 = lane&31
    // so for fixed lane the global stride per chunk is 2*C*4 = 4096 B and the
    // LDS stride is 64*16 = 1024 B.
    const uint64_t lane_gbase =
        xbase + 4ull * (((uint64_t)b * T_TOTAL + (uint64_t)(lane >> 5)) * C_TOTAL
                        + (uint64_t)(c0 + 4 * (lane & 31)));
    const uint32_t lane_lbase = lds_base + (uint32_t)lane * 16u;

    auto issue_tile = [&](int t0, int bufsel) {
        uint64_t ga = lane_gbase + (uint64_t)((uint32_t)t0 * (C_TOTAL * 4u));
        uint32_t la = lane_lbase + (uint32_t)bufsel * (TT * CC * 4u);
        #pragma unroll
        for (int k = 0; k < CHUNKS_PER_THREAD; ++k) {
            asm volatile("global_load_async_to_lds_b128 %0, %1, off"
                         :: "v"(la), "v"(ga)
                         : "memory");
            ga += 2u * C_TOTAL * 4u;   // +4096 B: two tile rows down
            la += THREADS * 16u;       // +1024 B
        }
    };

    // Prime the pipeline: tiles 0 and 1 in flight (32 outstanding per wave).
    issue_tile(0, 0);
    issue_tile(TT, 1);

    float* op = out + seq_base;            // walks out[b,t,c], stride C floats
    const int NT = T_TOTAL / TT;           // 128 tiles

    int cbuf = 0;   // buffer holding tile ti
    int ibuf = 2;   // buffer to receive tile ti+2
    for (int ti = 0; ti < NT; ++ti) {
        if (ti + 2 < NT) {
            issue_tile((ti + 2) * TT, ibuf);           // keep 48 in flight
            asm volatile("s_wait_asynccnt 32" ::: "memory");   // tile ti landed
        } else if (ti + 1 < NT) {
            asm volatile("s_wait_asynccnt 16" ::: "memory");
        } else {
            asm volatile("s_wait_asynccnt 0" ::: "memory");
        }
        __syncthreads();   // all waves' DMA for tile ti visible in LDS

        const float* buf = &tile[cbuf][0];

        #pragma unroll
        for (int tt = 0; tt < TT; ++tt) {
            const v2f xv = *(const v2f*)&buf[tt * CC + 2 * lane];
            const float r0 = xv.x - y0;
            const float r1 = xv.y - y1;
            const float g0 = __builtin_amdgcn_fmed3f(r0, -1.0f, 1.0f);
            const float g1 = __builtin_amdgcn_fmed3f(r1, -1.0f, 1.0f);
            y0 = fmaf(oma0, g0, y0);
            y1 = fmaf(oma1, g1, y1);
            v2f yv;
            yv.x = y0;
            yv.y = y1;
            __builtin_nontemporal_store(yv, (v2f*)op);
            op += C_TOTAL;
        }
        __syncthreads();   // buffer may be overwritten by next iteration's DMA

        cbuf = (cbuf == NBUF - 1) ? 0 : cbuf + 1;
        ibuf = (ibuf == NBUF - 1) ? 0 : ibuf + 1;
    }
}

extern "C" void kernel_launch(void* const* d_in, const int* in_sizes, int n_in,
                              void* d_out, int out_size, void* d_ws, size_t ws_size,
                              hipStream_t stream) {
    (void)in_sizes; (void)n_in; (void)out_size; (void)d_ws; (void)ws_size;
    const float* x           = (const float*)d_in[0];
    const float* logit_alpha = (const float*)d_in[1];
    float* out               = (float*)d_out;

    dim3 grid(B_TOTAL * (C_TOTAL / CC));   // 128 blocks
    dim3 block(THREADS);                   // 64 threads = 2 waves (wave32)
    hipLaunchKernelGGL(huber_ema_scan_kernel, grid, block, 0, stream,
                       x, logit_alpha, out);
}